// Phi_e_39470749450672
// MI455X (gfx1250) — compile-verified
//
#include <hip/hip_runtime.h>

typedef float v2f __attribute__((ext_vector_type(2)));
typedef float v8f __attribute__((ext_vector_type(8)));

#define N_NODES 1024
#define F_IN    32
#define HID     64
#define OUTF    32

// fast swish: x * sigmoid(x) with raw v_exp_f32 + v_rcp_f32 (no IEEE div sequence)
__device__ __forceinline__ float swishf(float x) {
    const float t = __builtin_amdgcn_exp2f(x * -1.44269504088896340736f); // e^-x
    return x * __builtin_amdgcn_rcpf(1.0f + t);
}

// ---- tiny precompute: A = Emb @ W1[:, :F]^T, B = Emb @ W1[:, F:]^T  (each 1024x64) ----
__global__ void precompute_AB(const float* __restrict__ Emb,
                              const float* __restrict__ W1,
                              float* __restrict__ Arow,
                              float* __restrict__ Brow) {
    const int n = blockIdx.x;      // node
    const int h = threadIdx.x;     // hidden unit, 0..63
    const float* e  = Emb + n * F_IN;
    const float* w1 = W1  + h * (2 * F_IN);
    float sa = 0.f, sb = 0.f;
#pragma unroll
    for (int f = 0; f < F_IN; ++f) {
        const float ev = e[f];
        sa = fmaf(ev, w1[f],        sa);
        sb = fmaf(ev, w1[F_IN + f], sb);
    }
    Arow[n * HID + h] = sa;
    Brow[n * HID + h] = sb;
}

// ---- main kernel: one block per i, one wave per 16-wide j tile stream ----
__global__ __launch_bounds__(256) void egnn_edge_mlp(
    const float* __restrict__ Edges,
    const float* __restrict__ Coords,
    const float* __restrict__ Arow,
    const float* __restrict__ Brow,
    const float* __restrict__ b1,
    const float* __restrict__ W2,
    const float* __restrict__ b2,
    float* __restrict__ Out)
{
    const int i    = blockIdx.x;
    const int lane = threadIdx.x & 31;
    const int wave = threadIdx.x >> 5;
    const int m    = lane & 15;   // row-in-tile (A) / col (B,C,D)
    const int khi  = lane >> 4;   // which K half of the 4-wide chunk

    // ---- hoisted per-i / per-wave register tiles ----
    v2f a_i[16];        // Arow[i][4kk + 2khi + {0,1}]
    v2f b1v[16];        // b1   [4kk + 2khi + {0,1}]
    v2f w2r[2][16];     // W2 as WMMA B-matrix chunks, 2 output tiles
    const float* arI = Arow + i * HID;
#pragma unroll
    for (int kk = 0; kk < 16; ++kk) {
        const int k0 = 4 * kk + 2 * khi;
        a_i[kk]    = *(const v2f*)(arI + k0);
        b1v[kk]    = *(const v2f*)(b1 + k0);
        w2r[0][kk] = *(const v2f*)(W2 + (m)      * HID + k0);
        w2r[1][kk] = *(const v2f*)(W2 + (16 + m) * HID + k0);
    }
    const float b2v0 = b2[m];
    const float b2v1 = b2[16 + m];

    const float cx = Coords[i * 3 + 0];
    const float cy = Coords[i * 3 + 1];
    const float cz = Coords[i * 3 + 2];

    for (int s = 0; s < 8; ++s) {
        const int j0 = (wave + 8 * s) * 16;
        const int j  = j0 + m;

        // w[i,j] = Edges[i,j] * ||c_i - c_j||  (zero-safe on diagonal)
        const float dx = cx - Coords[j * 3 + 0];
        const float dy = cy - Coords[j * 3 + 1];
        const float dz = cz - Coords[j * 3 + 2];
        const float sq = fmaf(dx, dx, fmaf(dy, dy, dz * dz));
        const float dist = sq > 0.f ? __builtin_amdgcn_sqrtf(sq) : 0.f;
        const float w = Edges[(size_t)i * N_NODES + j] * dist;
        const v2f wv = { w, w };

        v8f acc0 = {};
        v8f acc1 = {};
        const float* brJ = Brow + j * HID;
#pragma unroll
        for (int kk = 0; kk < 16; ++kk) {
            const int k0 = 4 * kk + 2 * khi;
            const v2f bb = *(const v2f*)(brJ + k0);
            // packed: pre1 = w*(a_i + b_j) + b1  -> v_pk_add_f32 + v_pk_fma_f32
            const v2f pre = (a_i[kk] + bb) * wv + b1v[kk];
            v2f h1;
            h1.x = swishf(pre.x);
            h1.y = swishf(pre.y);
            // D = A(16x4 f32) * B(4x16 f32) + C  — chained K accumulation
            acc0 = __builtin_amdgcn_wmma_f32_16x16x4_f32(
                false, h1, false, w2r[0][kk], (short)0, acc0, false, false);
            acc1 = __builtin_amdgcn_wmma_f32_16x16x4_f32(
                false, h1, false, w2r[1][kk], (short)0, acc1, false, false);
        }

        // epilogue: + b2, swish, store (16 j-rows x 32 outputs)
        float* outBase = Out + ((size_t)i * N_NODES + j0) * OUTF;
#pragma unroll
        for (int r = 0; r < 8; ++r) {
            const int jl = r + 8 * khi;               // C/D row for this lane half
            outBase[(size_t)jl * OUTF + m]      = swishf(acc0[r] + b2v0);
            outBase[(size_t)jl * OUTF + 16 + m] = swishf(acc1[r] + b2v1);
        }
    }
}

extern "C" void kernel_launch(void* const* d_in, const int* in_sizes, int n_in,
                              void* d_out, int out_size, void* d_ws, size_t ws_size,
                              hipStream_t stream) {
    const float* Edges  = (const float*)d_in[0];   // (N,N)
    const float* Coords = (const float*)d_in[1];   // (N,3)
    const float* Emb    = (const float*)d_in[2];   // (N,F)
    const float* W1     = (const float*)d_in[3];   // (HID, 2F)
    const float* b1     = (const float*)d_in[4];   // (HID,)
    const float* W2     = (const float*)d_in[5];   // (OUT, HID)
    const float* b2     = (const float*)d_in[6];   // (OUT,)
    float* Out = (float*)d_out;                    // (N,N,OUT)

    float* Arow = (float*)d_ws;                    // 256 KB
    float* Brow = Arow + (size_t)N_NODES * HID;    // 256 KB

    precompute_AB<<<N_NODES, HID, 0, stream>>>(Emb, W1, Arow, Brow);
    egnn_edge_mlp<<<N_NODES, 256, 0, stream>>>(Edges, Coords, Arow, Brow,
                                               b1, W2, b2, Out);
}